// ContextualizedNN_24541443130021
// MI455X (gfx1250) — compile-verified
//
#include <hip/hip_runtime.h>
#include <math.h>

// ---------------------------------------------------------------------------
// ContextualizedNN forward for MI455X (gfx1250, wave32, WMMA + async/TDM).
//
// Per batch b (one 128-thread / 4-wave32 workgroup per batch element):
//   scored_u = gather(scr_u)[50x50] @ gather(emb_u)[50x128]   (f32 WMMA)
//   scored_i = gather(scr_i)[50x50] @ gather(emb_i)[50x128]   (f32 WMMA)
//   inter    = scored_u * scored_i                             (VALU, in-reg)
//   h        = inter[64x128] @ w1[128x12] + b1                 (f32 WMMA)
//   out[b]   = mean_k sigmoid(h[k,:12] . w2 + b2)              (VALU + LDS)
//
// Data movement: embedding rows arrive via GLOBAL_LOAD_ASYNC_TO_LDS_B128
// (ASYNCcnt); w1 is staged by the Tensor Data Mover (TENSOR_LOAD_TO_LDS,
// TENSORcnt). Fallbacks are plain vectorized loads if builtins are absent.
// ---------------------------------------------------------------------------

typedef __attribute__((ext_vector_type(2))) float        v2f;
typedef __attribute__((ext_vector_type(8))) float        v8f;
typedef __attribute__((ext_vector_type(4))) unsigned int v4u;
typedef __attribute__((ext_vector_type(4))) int          v4i;
typedef __attribute__((ext_vector_type(8))) int          v8i;

typedef __attribute__((address_space(1))) v4i* gv4i_p;   // global v4i*
typedef __attribute__((address_space(3))) v4i* lv4i_p;   // LDS v4i*

#if __has_builtin(__builtin_amdgcn_global_load_async_to_lds_b128)
#define HAVE_ASYNC_LDS 1
#endif
#if __has_builtin(__builtin_amdgcn_tensor_load_to_lds) && \
    __has_builtin(__builtin_amdgcn_s_wait_tensorcnt)
#define HAVE_TDM 1
#endif

#define KN       50     // neighbors per node
#define DD       128    // embedding dim
#define HH       12     // hidden dim
#define MP       64     // padded row count (4 x 16 tiles)
#define KDP      52     // padded contraction dim (13 x 4 WMMA k-steps)
#define ASTRIDE  52     // LDS stride of score rows (m*52 % 64 distinct -> no conflicts)
#define ISTRIDE  132    // LDS stride of interaction rows (avoid 128-stride conflicts)

// LDS layout (float offsets). Score+embedding region is reused per side and
// then reused again for the interaction matrix.
#define OFF_A     0                       // [64][52]   = 3328 f
#define OFF_E     (OFF_A + MP*ASTRIDE)    // [52][128]  = 6656 f  (ends 9984; 16B aligned)
#define OFF_INTER 0                       // [64][132]  = 8448 f  (reuse A+E)
#define OFF_W1    9984                    // [128][12]  = 1536 f  (unpadded)
#define OFF_H     (OFF_W1 + DD*HH)        // [64][16]   = 1024 f
#define OFF_RED   (OFF_H + MP*16)         // [64]
#define SMEM_F    (OFF_RED + MP)          // 12608 floats = 50432 B

__global__ __launch_bounds__(128)
void cnn_fwd_kernel(const int*   __restrict__ user_idxs,
                    const int*   __restrict__ item_idxs,
                    const int*   __restrict__ user_idx_tensor,
                    const int*   __restrict__ item_idx_tensor,
                    const float* __restrict__ user_scr,
                    const float* __restrict__ item_scr,
                    const float* __restrict__ user_emb,
                    const float* __restrict__ item_emb,
                    const float* __restrict__ w1,
                    const float* __restrict__ b1,
                    const float* __restrict__ w2,
                    const float* __restrict__ b2,
                    float*       __restrict__ out)
{
    __shared__ float smem[SMEM_F];
    __shared__ int   sN[MP];

    const int tid   = threadIdx.x;
    const int lane  = tid & 31;
    const int wave  = tid >> 5;
    const int b     = blockIdx.x;

    const int mloc  = lane & 15;   // row/col-in-tile per fp32 WMMA layout
    const int khalf = lane >> 4;   // selects K pair {0,1} vs {2,3} of a k-step

    // ---- stage w1 [128][12] into LDS -----------------------------------
#ifdef HAVE_TDM
    if (wave == 0) {
        // Tensor DMA descriptor (ISA ch.8): 2D tile == whole 12x128 f32 tensor.
        const unsigned long long ga = (unsigned long long)(uintptr_t)w1;
        const unsigned lds_addr = (unsigned)(unsigned long long)
            (__attribute__((address_space(3))) float*)&smem[OFF_W1];
        v4u g0;
        g0.x = 1u;                                   // count=1, user mode
        g0.y = lds_addr;                             // lds_addr[31:0]
        g0.z = (unsigned)(ga & 0xffffffffu);         // global_addr[31:0]
        g0.w = (unsigned)((ga >> 32) & 0x01ffffffu)  // global_addr[56:32]
             | 0x80000000u;                          // type=2 ("image")
        v8i g1;
        g1[0] = 0x00020000;          // workgroup_mask=0, data_size=2 (4B)
        g1[1] = HH << 16;            // tensor_dim0 = 12
        g1[2] = DD << 16;            // tensor_dim1 = 128
        g1[3] = HH << 16;            // tile_dim0 = 12
        g1[4] = DD;                  // tile_dim1 = 128, tile_dim2 = 0
        g1[5] = HH;                  // tensor_dim0_stride = 12
        g1[6] = (int)((unsigned)(HH * DD) << 16); // tensor_dim1_stride[15:0]
        g1[7] = 0;
        v4i g2 = {0, 0, 0, 0};
        v4i g3 = {0, 0, 0, 0};
        v8i g4 = {0, 0, 0, 0, 0, 0, 0, 0};
        __builtin_amdgcn_tensor_load_to_lds(g0, g1, g2, g3, g4, 0);
        __builtin_amdgcn_s_wait_tensorcnt(0);
    }
#else
    for (int idx = tid; idx < DD * HH; idx += 128)
        smem[OFF_W1 + idx] = w1[idx];
#endif

    v8f accU[4][2], accI[4][2];
    const v8f vzero = {};

    // ---- two scored GEMMs: side 0 = user, side 1 = item -----------------
    for (int side = 0; side < 2; ++side) {
        const int*   idxs = side ? item_idxs       : user_idxs;
        const int*   idxt = side ? item_idx_tensor : user_idx_tensor;
        const float* scr  = side ? item_scr        : user_scr;
        const float* emb  = side ? item_emb        : user_emb;

        const int node = idxs[b];

        if (tid < MP) sN[tid] = (tid < KN) ? idxt[node * KN + tid] : 0;
        __syncthreads();

        // gather score rows -> [64][52] zero padded, float2 (rows 8B aligned)
        for (int idx = tid; idx < MP * 26; idx += 128) {
            const int r  = idx / 26;
            const int c2 = idx - r * 26;
            float2 v = float2{0.0f, 0.0f};
            if (r < KN && c2 < 25)
                v = *(const float2*)(scr + (size_t)sN[r] * KN + 2 * c2);
            *(float2*)&smem[OFF_A + r * ASTRIDE + 2 * c2] = v;
        }

        // gather embedding rows -> [52][128]; one row = 32 lanes x float4.
        // Rows are 512B aligned in global, 16B aligned in LDS.
        for (int j = wave; j < KDP; j += 4) {
            float* dst = &smem[OFF_E + j * DD + lane * 4];
            if (j < KN) {
                const float* src = emb + (size_t)sN[j] * DD + lane * 4;
#ifdef HAVE_ASYNC_LDS
                __builtin_amdgcn_global_load_async_to_lds_b128(
                    (gv4i_p)(uintptr_t)src, (lv4i_p)dst, 0, 0);
#else
                *(float4*)dst = *(const float4*)src;
#endif
            } else {
                *(float4*)dst = float4{0.0f, 0.0f, 0.0f, 0.0f};
            }
        }
#ifdef HAVE_ASYNC_LDS
#if __has_builtin(__builtin_amdgcn_s_wait_asynccnt)
        __builtin_amdgcn_s_wait_asynccnt(0);
#else
        asm volatile("s_wait_asynccnt 0" ::: "memory");
#endif
#endif
        __syncthreads();

        const int n0 = wave * 32;              // this wave's 32-column slice

        #pragma unroll
        for (int mt = 0; mt < 4; ++mt)
            #pragma unroll
            for (int nt = 0; nt < 2; ++nt)
                (side ? accI : accU)[mt][nt] = vzero;

        for (int kb = 0; kb < KDP; kb += 4) {
            // A fragments: lane {khalf,mloc} holds A[m][kb+2*khalf + {0,1}]
            v2f afrag[4];
            #pragma unroll
            for (int mt = 0; mt < 4; ++mt) {
                const int m = mt * 16 + mloc;
                afrag[mt].x = smem[OFF_A + m * ASTRIDE + kb + 2 * khalf];
                afrag[mt].y = smem[OFF_A + m * ASTRIDE + kb + 2 * khalf + 1];
            }
            #pragma unroll
            for (int nt = 0; nt < 2; ++nt) {
                const int n = n0 + nt * 16 + mloc;
                v2f bfrag;
                bfrag.x = smem[OFF_E + (kb + 2 * khalf)     * DD + n];
                bfrag.y = smem[OFF_E + (kb + 2 * khalf + 1) * DD + n];
                #pragma unroll
                for (int mt = 0; mt < 4; ++mt) {
                    v8f& acc = (side ? accI : accU)[mt][nt];
                    acc = __builtin_amdgcn_wmma_f32_16x16x4_f32(
                        false, afrag[mt], false, bfrag,
                        (short)0, acc, false, false);
                }
            }
        }
        __syncthreads();   // all waves done with A/E before reuse
    }

    // ---- interaction product (C-fragments match elementwise) -> LDS -----
    {
        const int n0 = wave * 32;
        #pragma unroll
        for (int mt = 0; mt < 4; ++mt) {
            const int mrow = mt * 16 + 8 * khalf;  // C layout: VGPR r -> M=mrow+r
            #pragma unroll
            for (int nt = 0; nt < 2; ++nt) {
                const int ncol = n0 + nt * 16 + mloc;
                const v8f u  = accU[mt][nt];
                const v8f iv = accI[mt][nt];
                #pragma unroll
                for (int r = 0; r < 8; ++r)
                    smem[OFF_INTER + (mrow + r) * ISTRIDE + ncol] = u[r] * iv[r];
            }
        }
    }
    __syncthreads();

    // ---- h = inter[64x128] @ w1[128x12]; one 16x16 tile per wave --------
    v8f hacc = vzero;
    {
        const int  m      = wave * 16 + mloc;
        const bool incol  = (mloc < HH);
        for (int kb = 0; kb < DD; kb += 4) {
            v2f a, bb;
            a.x  = smem[OFF_INTER + m * ISTRIDE + kb + 2 * khalf];
            a.y  = smem[OFF_INTER + m * ISTRIDE + kb + 2 * khalf + 1];
            bb.x = incol ? smem[OFF_W1 + (kb + 2 * khalf)     * HH + mloc] : 0.0f;
            bb.y = incol ? smem[OFF_W1 + (kb + 2 * khalf + 1) * HH + mloc] : 0.0f;
            hacc = __builtin_amdgcn_wmma_f32_16x16x4_f32(
                false, a, false, bb, (short)0, hacc, false, false);
        }
    }
    // store h + b1 (padded cols 12..15 get bias 0, never read)
    {
        const float bias = (mloc < HH) ? b1[mloc] : 0.0f;
        const int mrow = wave * 16 + 8 * khalf;
        #pragma unroll
        for (int r = 0; r < 8; ++r)
            smem[OFF_H + (mrow + r) * 16 + mloc] = hacc[r] + bias;
    }
    __syncthreads();

    // ---- logits, sigmoid, mean over valid K rows ------------------------
    if (tid < MP) {
        float v = 0.0f;
        if (tid < KN) {
            float logit = b2[0];
            #pragma unroll
            for (int n = 0; n < HH; ++n)
                logit += smem[OFF_H + tid * 16 + n] * w2[n];
            v = 1.0f / (1.0f + __expf(-logit));
        }
        smem[OFF_RED + tid] = v;
    }
    __syncthreads();
    #pragma unroll
    for (int s = 32; s > 0; s >>= 1) {
        if (tid < s) smem[OFF_RED + tid] += smem[OFF_RED + tid + s];
        __syncthreads();
    }
    if (tid == 0) out[b] = smem[OFF_RED] * (1.0f / (float)KN);
}

extern "C" void kernel_launch(void* const* d_in, const int* in_sizes, int n_in,
                              void* d_out, int out_size, void* d_ws, size_t ws_size,
                              hipStream_t stream)
{
    const int*   user_idxs       = (const int*)  d_in[0];
    const int*   item_idxs       = (const int*)  d_in[1];
    const int*   user_idx_tensor = (const int*)  d_in[2];
    const int*   item_idx_tensor = (const int*)  d_in[3];
    const float* user_scr        = (const float*)d_in[4];
    const float* item_scr        = (const float*)d_in[5];
    const float* user_emb        = (const float*)d_in[6];
    const float* item_emb        = (const float*)d_in[7];
    const float* w1              = (const float*)d_in[8];
    const float* b1              = (const float*)d_in[9];
    const float* w2              = (const float*)d_in[10];
    const float* b2              = (const float*)d_in[11];
    float*       out             = (float*)d_out;

    const int B = in_sizes[0];   // 4096 batch elements

    cnn_fwd_kernel<<<B, 128, 0, stream>>>(
        user_idxs, item_idxs, user_idx_tensor, item_idx_tensor,
        user_scr, item_scr, user_emb, item_emb,
        w1, b1, w2, b2, out);
}